// Mamba_TimeVariant_73443940761737
// MI455X (gfx1250) — compile-verified
//
#include <hip/hip_runtime.h>
#include <cstddef>

#define D_MODEL  1024
#define D_INNER  2048
#define D_STATE  16
#define D_CONV   4
#define DT_RANK  64
#define XZ_LD    (2 * D_INNER)           // 4096
#define XDBL_LD  (DT_RANK + 2 * D_STATE) // 96

typedef float v2f __attribute__((ext_vector_type(2)));
typedef float v8f __attribute__((ext_vector_type(8)));
typedef int   v4i __attribute__((ext_vector_type(4)));

#if __has_builtin(__builtin_amdgcn_global_load_async_to_lds_b128) && \
    __has_builtin(__builtin_amdgcn_s_wait_asynccnt)
#define USE_ASYNC_LDS 1
#else
#define USE_ASYNC_LDS 0
#endif

typedef __attribute__((address_space(1))) v4i* gptr_v4i;
typedef __attribute__((address_space(3))) v4i* lptr_v4i;

__device__ __forceinline__ float silu_f(float x) {
    return x / (1.0f + __expf(-x));
}
__device__ __forceinline__ float softplus_f(float x) {
    return (x > 20.0f) ? x : log1pf(__expf(x));
}

// Copy 16B global -> LDS (async on gfx1250, else through VGPRs).
__device__ __forceinline__ void cp_g2l_b128(const float* g, float* l) {
#if USE_ASYNC_LDS
    __builtin_amdgcn_global_load_async_to_lds_b128(
        (gptr_v4i)g, (lptr_v4i)l, /*offset=*/0, /*cpol=*/0);
#else
    *(float4*)l = *(const float4*)g;
#endif
}
__device__ __forceinline__ void cp_wait() {
#if USE_ASYNC_LDS
    __builtin_amdgcn_s_wait_asynccnt(0);
#endif
}

// ---------------------------------------------------------------------------
// Blocked fp32 WMMA GEMM:  C[M,N] = A[M,K] * W[N,K]^T   (+ optional epilogue)
//
// Block = 256 threads (8 waves).  Block tile 128x64, k-chunk KC=32, double-
// buffered LDS staging (async-to-LDS on gfx1250).  Wave w computes rows
// [w*16, w*16+16) of the block tile across all 64 columns (4 accumulators).
//
// Requires: M%128==0, N%64==0, K%32==0 (true for all call sites here).
// V_WMMA_F32_16X16X4_F32 fragments:
//   A 16x4 f32: lane m=lane&15 holds A[m][k0+2*half], A[m][k0+2*half+1]
//   B mirrors it on W rows (B = W^T);  C/D: VGPR r -> row r+8*half, col lane&15
// ACT==1: v = softplus(v + bias[col])
// ---------------------------------------------------------------------------
#define KC     32
#define LDSPAD 36   // row stride in floats: 16B-aligned rows, conflict-free b64 reads

template <int ACT>
__global__ void __launch_bounds__(256)
wmma_gemm_blocked_kernel(const float* __restrict__ A,
                         const float* __restrict__ W,
                         float* __restrict__ C,
                         const float* __restrict__ bias,
                         int M, int N, int K,
                         int lda, int ldw, int ldc)
{
    __shared__ float As[2][128][LDSPAD];
    __shared__ float Ws[2][64][LDSPAD];

    const int tid  = threadIdx.x;
    const int lane = tid & 31;
    const int wave = tid >> 5;
    const int half = lane >> 4;   // selects K pair {0,1} vs {2,3}
    const int ml   = lane & 15;   // A row / B col within 16x16 tile

    const int nBlocksN = N / 64;
    const int bm = blockIdx.x / nBlocksN;
    const int bn = blockIdx.x % nBlocksN;

    const float* Ablk = A + (size_t)bm * 128 * lda;
    const float* Wblk = W + (size_t)bn * 64 * ldw;

    const int nChunks = K / KC;

    // Stage one k-chunk (A: 128x32 = 1024 float4 ops; W: 64x32 = 512) across
    // 256 threads: 4 + 2 float4 copies per thread.
    auto stage = [&](int buf, int k0) {
#pragma unroll
        for (int i = 0; i < 4; ++i) {
            const int item = tid + i * 256;       // 0..1023
            const int r = item >> 3, s = item & 7;
            cp_g2l_b128(Ablk + (size_t)r * lda + k0 + s * 4,
                        &As[buf][r][s * 4]);
        }
#pragma unroll
        for (int i = 0; i < 2; ++i) {
            const int item = tid + i * 256;       // 0..511
            const int r = item >> 3, s = item & 7;
            cp_g2l_b128(Wblk + (size_t)r * ldw + k0 + s * 4,
                        &Ws[buf][r][s * 4]);
        }
    };

    v8f acc[4];
#pragma unroll
    for (int j = 0; j < 4; ++j) acc[j] = (v8f)(0.0f);

    stage(0, 0);

    for (int c = 0; c < nChunks; ++c) {
        const int buf = c & 1;
        cp_wait();            // my share of chunk c's copies landed in LDS
        __syncthreads();      // everyone's share landed; prev reads of buf done

        if (c + 1 < nChunks) stage(buf ^ 1, (c + 1) * KC);

        const float* arow = &As[buf][wave * 16 + ml][2 * half];
#pragma unroll
        for (int kk = 0; kk < KC; kk += 4) {
            const v2f a = *(const v2f*)(arow + kk);
#pragma unroll
            for (int j = 0; j < 4; ++j) {
                const v2f b = *(const v2f*)(&Ws[buf][j * 16 + ml][kk + 2 * half]);
                acc[j] = __builtin_amdgcn_wmma_f32_16x16x4_f32(
                    false, a, false, b, (short)0, acc[j], false, false);
            }
        }
    }

#pragma unroll
    for (int j = 0; j < 4; ++j) {
        const int ocol = bn * 64 + j * 16 + ml;
#pragma unroll
        for (int r = 0; r < 8; ++r) {
            const int orow = bm * 128 + wave * 16 + r + 8 * half;
            float v = acc[j][r];
            if (ACT == 1) v = softplus_f(v + bias[ocol]);
            C[(size_t)orow * ldc + ocol] = v;
        }
    }
}

// ---------------------------------------------------------------------------
// Simple per-wave WMMA GEMM (one 16x16 tile per wave) for the narrow N=96
// x_proj GEMM.  Same fragment layout as above.
// ---------------------------------------------------------------------------
__global__ void wmma_gemm_narrow_kernel(const float* __restrict__ A,
                                        const float* __restrict__ W,
                                        float* __restrict__ C,
                                        int M, int N, int K,
                                        int lda, int ldw, int ldc)
{
    const int lane = threadIdx.x & 31;
    const int wave = threadIdx.x >> 5;
    const int half = lane >> 4;
    const int ml   = lane & 15;

    const int tilesN = N / 16;
    const int totalTiles = (M / 16) * tilesN;
    const int tile = blockIdx.x * (blockDim.x >> 5) + wave;
    if (tile >= totalTiles) return;  // wave-uniform

    const int tm = tile / tilesN;
    const int tn = tile % tilesN;

    const float* pA = A + (size_t)(tm * 16 + ml) * lda + half * 2;
    const float* pW = W + (size_t)(tn * 16 + ml) * ldw + half * 2;

    v8f acc = (v8f)(0.0f);
    for (int k = 0; k < K; k += 4) {
        const v2f a = *(const v2f*)(pA + k);
        const v2f b = *(const v2f*)(pW + k);
        acc = __builtin_amdgcn_wmma_f32_16x16x4_f32(
            false, a, false, b, (short)0, acc, false, false);
    }
#pragma unroll
    for (int r = 0; r < 8; ++r)
        C[(size_t)(tm * 16 + r + 8 * half) * ldc + tn * 16 + ml] = acc[r];
}

// ---------------------------------------------------------------------------
// Depthwise causal conv (K=4) + bias + SiLU.
// ---------------------------------------------------------------------------
__global__ void conv_silu_kernel(const float* __restrict__ xz,
                                 const float* __restrict__ conv_w,
                                 const float* __restrict__ conv_b,
                                 float* __restrict__ u,
                                 int L, int total)
{
    const int idx = blockIdx.x * blockDim.x + threadIdx.x;
    if (idx >= total) return;
    const int d = idx & (D_INNER - 1);
    const int t = (idx / D_INNER) % L;
    const int b = idx / (D_INNER * L);

    const float4 w = *(const float4*)(conv_w + (size_t)d * D_CONV);
    const size_t base = ((size_t)b * L + t) * XZ_LD + d;

    float acc = conv_b[d];
    acc += (t >= 3 ? xz[base - 3 * (size_t)XZ_LD] : 0.0f) * w.x;
    acc += (t >= 2 ? xz[base - 2 * (size_t)XZ_LD] : 0.0f) * w.y;
    acc += (t >= 1 ? xz[base - 1 * (size_t)XZ_LD] : 0.0f) * w.z;
    acc += xz[base] * w.w;

    u[idx] = silu_f(acc);
}

// ---------------------------------------------------------------------------
// Selective scan: one thread per (b,d) channel, h[16] in VGPRs, L sequential
// steps.  delta_y: softplus(delta) in, y*silu(z) out (in place, race-free).
// ---------------------------------------------------------------------------
__global__ void scan_kernel(const float* __restrict__ xz,
                            const float* __restrict__ u,
                            const float* __restrict__ xdbl,
                            float* __restrict__ delta_y,
                            const float* __restrict__ A_log,
                            const float* __restrict__ Dp,
                            int L)
{
    const int chan = blockIdx.x * blockDim.x + threadIdx.x; // 0..B*D_INNER-1
    const int d = chan & (D_INNER - 1);
    const int b = chan >> 11; // D_INNER == 2048

    float Ar[D_STATE];
#pragma unroll
    for (int n = 0; n < D_STATE; ++n)
        Ar[n] = -__expf(A_log[(size_t)d * D_STATE + n]);

    float h[D_STATE];
#pragma unroll
    for (int n = 0; n < D_STATE; ++n) h[n] = 0.0f;

    const float Dd = Dp[d];

    for (int t = 0; t < L; ++t) {
        const size_t bt = (size_t)b * L + t;
        const float dt = delta_y[bt * D_INNER + d];
        const float uu = u[bt * D_INNER + d];

        const float4* p4 = (const float4*)(xdbl + bt * XDBL_LD + DT_RANK);
        float Bv[D_STATE], Cv[D_STATE];
#pragma unroll
        for (int i = 0; i < 4; ++i) {
            const float4 vb = p4[i];
            Bv[4 * i + 0] = vb.x; Bv[4 * i + 1] = vb.y;
            Bv[4 * i + 2] = vb.z; Bv[4 * i + 3] = vb.w;
            const float4 vc = p4[i + 4];
            Cv[4 * i + 0] = vc.x; Cv[4 * i + 1] = vc.y;
            Cv[4 * i + 2] = vc.z; Cv[4 * i + 3] = vc.w;
        }

        const float du = dt * uu;
        float y = 0.0f;
#pragma unroll
        for (int n = 0; n < D_STATE; ++n) {
            h[n] = h[n] * __expf(dt * Ar[n]) + du * Bv[n];
            y += h[n] * Cv[n];
        }

        const float zz = xz[bt * XZ_LD + D_INNER + d];
        delta_y[bt * D_INNER + d] = (y + Dd * uu) * silu_f(zz);
    }
}

// ---------------------------------------------------------------------------
extern "C" void kernel_launch(void* const* d_in, const int* in_sizes, int n_in,
                              void* d_out, int out_size, void* d_ws, size_t ws_size,
                              hipStream_t stream)
{
    const float* hidden     = (const float*)d_in[0];
    const float* in_proj_w  = (const float*)d_in[1];
    const float* conv_w     = (const float*)d_in[2];
    const float* conv_b     = (const float*)d_in[3];
    const float* x_proj_w   = (const float*)d_in[4];
    const float* dt_proj_w  = (const float*)d_in[5];
    const float* dt_proj_b  = (const float*)d_in[6];
    const float* A_log      = (const float*)d_in[7];
    const float* Dp         = (const float*)d_in[8];
    const float* out_proj_w = (const float*)d_in[9];

    const int BL = in_sizes[0] / D_MODEL; // B * L = 8192
    const int B  = 4;
    const int L  = BL / B;

    float* xz    = (float*)d_ws;                 // BL * 4096
    float* u     = xz   + (size_t)BL * XZ_LD;    // BL * 2048
    float* xdbl  = u    + (size_t)BL * D_INNER;  // BL * 96
    float* delta = xdbl + (size_t)BL * XDBL_LD;  // BL * 2048 (y in place)

    const int TPB = 256;

    // 1) xz = hidden @ in_proj_w^T        (8192 x 4096 x 1024)
    wmma_gemm_blocked_kernel<0><<<(BL / 128) * (XZ_LD / 64), TPB, 0, stream>>>(
        hidden, in_proj_w, xz, nullptr,
        BL, XZ_LD, D_MODEL, D_MODEL, D_MODEL, XZ_LD);

    // 2) u = silu(causal_conv1d(x) + conv_b)
    {
        const int total = BL * D_INNER;
        conv_silu_kernel<<<(total + TPB - 1) / TPB, TPB, 0, stream>>>(
            xz, conv_w, conv_b, u, L, total);
    }

    // 3) x_dbl = u @ x_proj_w^T           (8192 x 96 x 2048)
    {
        const int tiles = (BL / 16) * (XDBL_LD / 16);
        wmma_gemm_narrow_kernel<<<(tiles + 7) / 8, TPB, 0, stream>>>(
            u, x_proj_w, xdbl, BL, XDBL_LD, D_INNER, D_INNER, D_INNER, XDBL_LD);
    }

    // 4) delta = softplus(dt_lo @ dt_proj_w^T + dt_proj_b)   (8192 x 2048 x 64)
    wmma_gemm_blocked_kernel<1><<<(BL / 128) * (D_INNER / 64), TPB, 0, stream>>>(
        xdbl, dt_proj_w, delta, dt_proj_b,
        BL, D_INNER, DT_RANK, XDBL_LD, DT_RANK, D_INNER);

    // 5) selective scan + gating; y overwrites delta in place
    scan_kernel<<<(B * D_INNER) / TPB, TPB, 0, stream>>>(
        xz, u, xdbl, delta, A_log, Dp, L);

    // 6) out = y @ out_proj_w^T           (8192 x 1024 x 2048)
    wmma_gemm_blocked_kernel<0><<<(BL / 128) * (D_MODEL / 64), TPB, 0, stream>>>(
        delta, out_proj_w, (float*)d_out, nullptr,
        BL, D_MODEL, D_INNER, D_INNER, D_INNER, D_MODEL);
}